// MKGC_60078002536517
// MI455X (gfx1250) — compile-verified
//
#include <hip/hip_runtime.h>
#include <hip/hip_bf16.h>

// GCN: out = sum_k relu( (D^-1/2 (A+I) D^-1/2) x W_k + b_k )
// Linearity rewrite: y = Ahat @ x computed once (scatter), then 3 dense 64x64
// GEMMs via V_WMMA_F32_16X16X4_F32 with LDS-staged pre-swizzled weights and a
// fused bias+relu+sum epilogue.

#define NNODES 50000
#define NEDGES 800000
#define DD     64
#define NK     3

typedef float v2f __attribute__((ext_vector_type(2)));
typedef float v8f __attribute__((ext_vector_type(8)));

// ---- 1) deg[i] = 1 (self loop) ----------------------------------------------
__global__ void gcn_init_deg(float* __restrict__ deg) {
    int i = blockIdx.x * blockDim.x + threadIdx.x;
    if (i < NNODES) deg[i] = 1.0f;
}

// ---- 2) deg[dst] += 1 per edge ----------------------------------------------
__global__ void gcn_edge_deg(const long long* __restrict__ dst, float* __restrict__ deg) {
    int e = blockIdx.x * blockDim.x + threadIdx.x;
    if (e < NEDGES) atomicAdd(&deg[(int)dst[e]], 1.0f);
}

// ---- 3) deg -> rsqrt(deg) in place (deg >= 1 always, so no zero guard) ------
__global__ void gcn_dinv(float* __restrict__ deg) {
    int i = blockIdx.x * blockDim.x + threadIdx.x;
    if (i < NNODES) deg[i] = rsqrtf(deg[i]);
}

// ---- 4) zero accumulator y --------------------------------------------------
__global__ void gcn_zero_y(float* __restrict__ y) {
    int i = blockIdx.x * blockDim.x + threadIdx.x;
    if (i < NNODES * DD) y[i] = 0.0f;
}

// ---- 5) y[dst] += dinv[src]*dinv[dst] * x[src]  (one (edge,dim) per thread) -
__global__ void gcn_scatter(const float* __restrict__ x,
                            const long long* __restrict__ src,
                            const long long* __restrict__ dst,
                            const float* __restrict__ dinv,
                            float* __restrict__ y) {
    long long idx = (long long)blockIdx.x * blockDim.x + threadIdx.x;
    int e = (int)(idx >> 6);
    int d = (int)(idx & 63);
    if (e < NEDGES) {
        int s = (int)src[e];
        int t = (int)dst[e];
        float nrm = dinv[s] * dinv[t];
        atomicAdd(&y[t * DD + d], nrm * x[s * DD + d]);
    }
}

// ---- 6) self loop: y[i] += dinv[i]^2 * x[i] ---------------------------------
__global__ void gcn_selfloop(const float* __restrict__ x,
                             const float* __restrict__ dinv,
                             float* __restrict__ y) {
    int i = blockIdx.x * blockDim.x + threadIdx.x;
    if (i < NNODES * DD) {
        float di = dinv[i >> 6];
        y[i] += di * di * x[i];
    }
}

// ---- 7) out = sum_k relu(y @ W_k + b_k) via WMMA f32 16x16x4 ----------------
// One wave computes a 16-row x 64-col strip. ISA VGPR layouts (CDNA5 ISA):
//   A (16x4 f32, 2 VGPRs):  lane = 16*Khi + M ; element j holds K = 4*kk + 2*(lane/16) + j
//   B (4x16 f32, 2 VGPRs):  lane = 16*Khi + N ; same K mapping
//   C/D (16x16 f32, 8 VGPRs): vgpr r, lane l -> row = r + 8*(l/16), col = l%16
// W is staged in LDS pre-swizzled so each lane's B fragment is one float2:
//   Wlds[((k*16 + kk)*4 + tile)*32 + lane] = {W[k][K0][n], W[k][K0+1][n]}
__global__ __launch_bounds__(256) void gcn_gemm_relu_sum(
        const float* __restrict__ z,      // y after self-loop: Ahat @ x, [N,64]
        const float* __restrict__ W,      // [3,64,64]
        const float* __restrict__ b,      // [3,64]
        float* __restrict__ out) {        // [N,64]
    __shared__ v2f Wlds[NK * 16 * 4 * 32];   // 48 KB, pre-swizzled B fragments

    // Cooperative fill: linear index i over [3,64,64] -> fragment position.
    float* Wf = (float*)Wlds;
    for (int i = threadIdx.x; i < NK * DD * DD; i += 256) {
        int k    = i >> 12;        // which weight matrix
        int r    = (i >> 6) & 63;  // K row
        int c    = i & 63;         // N col
        int kk   = r >> 2;         // K step (of 4)
        int half = (r >> 1) & 1;   // lane half holding this K
        int j    = r & 1;          // element within float2
        int tile = c >> 4;         // N tile
        int lane = half * 16 + (c & 15);
        Wf[((((k * 16 + kk) * 4 + tile) * 32 + lane) << 1) + j] = W[i];
    }
    __syncthreads();

    const int lane = threadIdx.x & 31;
    const int wave = blockIdx.x * 8 + (threadIdx.x >> 5);
    const int m0   = wave * 16;
    if (m0 >= NNODES) return;             // wave-uniform: EXEC all-1s at WMMAs

    const int half = lane >> 4;           // 0: lanes 0-15, 1: lanes 16-31
    const int l16  = lane & 15;

    // Hoist this lane's A strip: 16 float2 covering K=0..63 (its half's pairs).
    const v2f* zrow2 = (const v2f*)(z + (m0 + l16) * DD);
    v2f afrag[16];
    #pragma unroll
    for (int kk = 0; kk < 16; ++kk) afrag[kk] = zrow2[kk * 2 + half];

    v8f sum0 = {}, sum1 = {}, sum2 = {}, sum3 = {};

    for (int k = 0; k < NK; ++k) {
        const v2f* wk = Wlds + k * 2048 + lane;   // + kk*128 + tile*32
        v8f acc0 = {}, acc1 = {}, acc2 = {}, acc3 = {};

        #pragma unroll
        for (int kk = 0; kk < 16; ++kk) {
            const v2f a  = afrag[kk];
            const v2f b0 = wk[kk * 128 +  0];
            const v2f b1 = wk[kk * 128 + 32];
            const v2f b2 = wk[kk * 128 + 64];
            const v2f b3 = wk[kk * 128 + 96];
            acc0 = __builtin_amdgcn_wmma_f32_16x16x4_f32(false, a, false, b0, (short)0, acc0, false, false);
            acc1 = __builtin_amdgcn_wmma_f32_16x16x4_f32(false, a, false, b1, (short)0, acc1, false, false);
            acc2 = __builtin_amdgcn_wmma_f32_16x16x4_f32(false, a, false, b2, (short)0, acc2, false, false);
            acc3 = __builtin_amdgcn_wmma_f32_16x16x4_f32(false, a, false, b3, (short)0, acc3, false, false);
        }

        const float bv0 = b[k * DD +  0 + l16];
        const float bv1 = b[k * DD + 16 + l16];
        const float bv2 = b[k * DD + 32 + l16];
        const float bv3 = b[k * DD + 48 + l16];
        #pragma unroll
        for (int r = 0; r < 8; ++r) {
            float v;
            v = acc0[r] + bv0; sum0[r] += v > 0.f ? v : 0.f;
            v = acc1[r] + bv1; sum1[r] += v > 0.f ? v : 0.f;
            v = acc2[r] + bv2; sum2[r] += v > 0.f ? v : 0.f;
            v = acc3[r] + bv3; sum3[r] += v > 0.f ? v : 0.f;
        }
    }

    #pragma unroll
    for (int r = 0; r < 8; ++r) {
        float* orow = out + (m0 + r + 8 * half) * DD + l16;
        orow[ 0] = sum0[r];
        orow[16] = sum1[r];
        orow[32] = sum2[r];
        orow[48] = sum3[r];
    }
}

extern "C" void kernel_launch(void* const* d_in, const int* in_sizes, int n_in,
                              void* d_out, int out_size, void* d_ws, size_t ws_size,
                              hipStream_t stream) {
    const float*     x  = (const float*)d_in[0];
    const long long* ei = (const long long*)d_in[1];   // int64 per reference
    const float*     W  = (const float*)d_in[2];
    const float*     b  = (const float*)d_in[3];
    float*           out = (float*)d_out;

    const long long* src = ei;            // edge_index[0]
    const long long* dst = ei + NEDGES;   // edge_index[1]

    // workspace layout: dinv [N] (padded to 50176), then y [N*64]
    float* dinv = (float*)d_ws;
    float* y    = dinv + 50176;

    const int T = 256;
    gcn_init_deg<<<(NNODES + T - 1) / T, T, 0, stream>>>(dinv);
    gcn_edge_deg<<<(NEDGES + T - 1) / T, T, 0, stream>>>(dst, dinv);
    gcn_dinv<<<(NNODES + T - 1) / T, T, 0, stream>>>(dinv);
    gcn_zero_y<<<(NNODES * DD + T - 1) / T, T, 0, stream>>>(y);

    long long scat_threads = (long long)NEDGES * DD;
    gcn_scatter<<<(unsigned)((scat_threads + T - 1) / T), T, 0, stream>>>(x, src, dst, dinv, y);
    gcn_selfloop<<<(NNODES * DD + T - 1) / T, T, 0, stream>>>(x, dinv, y);

    // 3125 row-tiles of 16; 8 waves (of 32) per 256-thread block
    const int waves_needed = (NNODES + 15) / 16;            // 3125
    const int blocks = (waves_needed + 7) / 8;              // 391
    gcn_gemm_relu_sum<<<blocks, T, 0, stream>>>(y, W, b, out);
}